// LocalAttention_60979945669071
// MI455X (gfx1250) — compile-verified
//
#include <hip/hip_runtime.h>

// Sliding-window causal additive mask:
//   out[b,h,i,j] = scores[b,h,i,j] + ((j >= i-128 && j <= i) ? 0 : -inf)
//
// Memory-bound streaming op (1 GiB in, 1 GiB out). Strategy:
//  * Only READ input inside the kept band [i-128, i] (~6.3% of columns):
//    everywhere else the result is -inf regardless of the input.
//    Traffic: 1 GiB stores + ~68 MB loads  =>  ~47 us @ 23.3 TB/s.
//  * 64 B per thread (4 x b128) to amortize index math so the kernel stays
//    bandwidth-bound, not VALU-issue-bound.
//  * Non-temporal B128 ops: one-pass stream, don't thrash the 192 MB L2.
//  * WMMA is inapplicable: zero matmul FLOPs in this op.

typedef float v4f __attribute__((ext_vector_type(4)));

#define WINDOW_HALF 128  // WINDOW_SIZE / 2

__global__ __launch_bounds__(256)
void local_causal_mask_kernel(const float* __restrict__ scores,
                              const int* __restrict__ seq_len_p,
                              float* __restrict__ out,
                              unsigned int nseg /* total elements / 16 */) {
    unsigned int sid = blockIdx.x * blockDim.x + threadIdx.x;
    if (sid >= nseg) return;

    // Uniform scalar load of S (s_load_b32); S == 2048, power of two.
    unsigned int S     = (unsigned int)*seq_len_p;
    unsigned int shift = 31u - (unsigned int)__clz(S);
    unsigned int m     = S - 1u;

    // Flat element index of this 16-float segment (2^28 max, fits u32).
    unsigned int idx = sid << 4;
    unsigned int j0  = idx & m;            // first column of segment
    unsigned int i   = (idx >> shift) & m; // row within the S x S tile

    const float NEG_INF = -__builtin_huge_valf();
    int lo = (int)i - WINDOW_HALF;         // kept band: lo <= j <= i
    int hi = (int)i;

    const float* __restrict__ src = scores + idx;
    float*       __restrict__ dst = out + idx;

    if ((int)(j0 + 15u) < lo || (int)j0 > hi) {
        // Segment fully masked: store -inf, never touch the input.
        v4f ninf = (v4f){NEG_INF, NEG_INF, NEG_INF, NEG_INF};
#pragma unroll
        for (int c = 0; c < 4; ++c)
            __builtin_nontemporal_store(ninf, (v4f*)(dst + 4 * c));
    } else if ((int)j0 >= lo && (int)(j0 + 15u) <= hi) {
        // Segment fully inside the band: pure streaming copy.
#pragma unroll
        for (int c = 0; c < 4; ++c) {
            v4f v = __builtin_nontemporal_load((const v4f*)(src + 4 * c));
            __builtin_nontemporal_store(v, (v4f*)(dst + 4 * c));
        }
    } else {
        // Border segment (<= 2 per row): per-element select.
#pragma unroll
        for (int c = 0; c < 4; ++c) {
            v4f v = __builtin_nontemporal_load((const v4f*)(src + 4 * c));
#pragma unroll
            for (int k = 0; k < 4; ++k) {
                int j = (int)j0 + 4 * c + k;
                if (j < lo || j > hi) v[k] = NEG_INF;
            }
            __builtin_nontemporal_store(v, (v4f*)(dst + 4 * c));
        }
    }
}

extern "C" void kernel_launch(void* const* d_in, const int* in_sizes, int n_in,
                              void* d_out, int out_size, void* d_ws, size_t ws_size,
                              hipStream_t stream) {
    const float* scores  = (const float*)d_in[0];
    const int*   seq_len = (const int*)d_in[1];
    float*       out     = (float*)d_out;

    // S*S is divisible by 16 (S = 2048), so no remainder handling needed.
    unsigned int nseg = (unsigned int)(in_sizes[0] / 16);
    const int threads = 256;  // 8 wave32s per block
    unsigned int blocks = (nseg + threads - 1) / threads;

    local_causal_mask_kernel<<<blocks, threads, 0, stream>>>(scores, seq_len, out, nseg);
}